// channel_grouping_layer_54735063220846
// MI455X (gfx1250) — compile-verified
//
#include <hip/hip_runtime.h>

typedef float v2f __attribute__((ext_vector_type(2)));
typedef float v8f __attribute__((ext_vector_type(8)));

#define B_   64
#define C_   2048
#define P_   4
#define S_   196      // 14*14
#define HID_ 4096
#define N2_  8192     // C*P

__device__ __forceinline__ float sigmoidf(float x) {
    return 1.0f / (1.0f + __expf(-x));
}

// ---------------------------------------------------------------------------
// Kernel 1: g[b][c] = mean over 196 spatial elements of x[b][c][:,:]
// One wave32 per (b,c) row; contiguous 784B read per wave; butterfly reduce.
// ---------------------------------------------------------------------------
__global__ void cg_pool_kernel(const float* __restrict__ x, float* __restrict__ g) {
    int wave = (int)((blockIdx.x * blockDim.x + threadIdx.x) >> 5);
    int lane = (int)(threadIdx.x & 31);
    if (wave >= B_ * C_) return;
    const float* row = x + (size_t)wave * S_;
    float s = 0.f;
    for (int i = lane; i < S_; i += 32) s += row[i];
#pragma unroll
    for (int off = 16; off >= 1; off >>= 1) s += __shfl_xor(s, off, 32);
    if (lane == 0) g[wave] = s * (1.0f / (float)S_);
}

// ---------------------------------------------------------------------------
// Kernels 2/3: OUT[m][n] = sigmoid( sum_k A[m][k]*Bw[n][k] + bias[n] )
// M = 64 rows (4 tiles of 16). One wave per 16x16 output tile.
// f32 WMMA 16x16x4 lane layout:
//   A frag: lane l -> m = l&15, a[v] = A[m][k + v + 2*(l>>4)]
//   B frag: lane l -> n = l&15, b[v] = Bw[n][k + v + 2*(l>>4)]   (Bw is N-major over K)
//   D:      lane l -> col n = l&15, row m = v + 8*(l>>4)
// ---------------------------------------------------------------------------
__global__ void cg_gemm_sig_kernel(const float* __restrict__ A,
                                   const float* __restrict__ Bw,
                                   const float* __restrict__ bias,
                                   float* __restrict__ OUT,
                                   int K, int NT) {
    int tile = (int)(blockIdx.x * (blockDim.x >> 5) + (threadIdx.x >> 5));
    int lane = (int)(threadIdx.x & 31);
    if (tile >= 4 * NT) return;          // wave-uniform
    int mtile = tile / NT;
    int ntile = tile - mtile * NT;
    int nl = lane & 15;
    int hh = lane >> 4;

    int m = mtile * 16 + nl;
    int n = ntile * 16 + nl;
    const float* arow = A  + (size_t)m * K;
    const float* brow = Bw + (size_t)n * K;

    v8f acc = {0.f, 0.f, 0.f, 0.f, 0.f, 0.f, 0.f, 0.f};
#pragma unroll 8
    for (int k = 0; k < K; k += 4) {
        v2f a = *(const v2f*)(arow + k + 2 * hh);
        v2f b = *(const v2f*)(brow + k + 2 * hh);
        acc = __builtin_amdgcn_wmma_f32_16x16x4_f32(false, a, false, b,
                                                    (short)0, acc, false, false);
    }

    int N = NT * 16;
    float bn = bias[n];
#pragma unroll
    for (int v = 0; v < 8; ++v) {
        int mm = mtile * 16 + v + 8 * hh;
        OUT[(size_t)mm * N + n] = sigmoidf(acc[v] + bn);
    }
}

// ---------------------------------------------------------------------------
// Kernel 4: feat[b][p][s] = (1/C) * sum_c Wg[b][p][c] * x[b][c][s]
// Per batch: [4 x 2048] @ [2048 x 196]. M=4 padded to 16, N=196 -> 13 tiles.
// B-fragment loads are fully coalesced (x row-major over s).
// ---------------------------------------------------------------------------
__global__ void cg_feat_kernel(const float* __restrict__ Wg,   // [B][P*C] row-major
                               const float* __restrict__ x,    // [B][C][S]
                               float* __restrict__ feat) {     // [B][P][S]
    const int NT = 13;                   // ceil(196/16)
    int tile = (int)(blockIdx.x * (blockDim.x >> 5) + (threadIdx.x >> 5));
    int lane = (int)(threadIdx.x & 31);
    if (tile >= B_ * NT) return;         // wave-uniform
    int b  = tile / NT;
    int nt = tile - b * NT;
    int nl = lane & 15;
    int hh = lane >> 4;

    int n = nt * 16 + nl;
    bool nok = (n < S_);
    bool mok = (nl < P_);

    const float* arow = Wg + (size_t)b * N2_ + (size_t)nl * C_;
    const float* xb   = x  + (size_t)b * C_ * S_;

    v8f acc = {0.f, 0.f, 0.f, 0.f, 0.f, 0.f, 0.f, 0.f};
#pragma unroll 4
    for (int k = 0; k < C_; k += 4) {
        v2f a = {0.f, 0.f};
        if (mok) a = *(const v2f*)(arow + k + 2 * hh);
        v2f bv = {0.f, 0.f};
        if (nok) {
            bv.x = xb[(size_t)(k + 2 * hh)     * S_ + n];
            bv.y = xb[(size_t)(k + 2 * hh + 1) * S_ + n];
        }
        acc = __builtin_amdgcn_wmma_f32_16x16x4_f32(false, a, false, bv,
                                                    (short)0, acc, false, false);
    }

    if (hh == 0 && nok) {
#pragma unroll
        for (int v = 0; v < P_; ++v)     // rows m = v (only m<4 valid)
            feat[(size_t)b * (P_ * S_) + (size_t)v * S_ + n] = acc[v] * (1.0f / (float)C_);
    }
}

// ---------------------------------------------------------------------------
extern "C" void kernel_launch(void* const* d_in, const int* in_sizes, int n_in,
                              void* d_out, int out_size, void* d_ws, size_t ws_size,
                              hipStream_t stream) {
    const float* x  = (const float*)d_in[0];   // [64,2048,14,14]
    const float* w1 = (const float*)d_in[1];   // [4096,2048]
    const float* b1 = (const float*)d_in[2];   // [4096]
    const float* w2 = (const float*)d_in[3];   // [8192,4096]
    const float* b2 = (const float*)d_in[4];   // [8192]

    float* g = (float*)d_ws;                   // [64,2048]   512 KB
    float* h = g + (size_t)B_ * C_;            // [64,4096]   1 MB

    float* w_out    = (float*)d_out;           // [64,4,2048] = [64,8192]
    float* feat_out = w_out + (size_t)B_ * N2_;// [64,4,196]

    // 1) global average pool: 131072 waves, 8 waves per 256-thread block
    cg_pool_kernel<<<(B_ * C_) / 8, 256, 0, stream>>>(x, g);

    // 2) h = sigmoid(g @ w1^T + b1): tiles = 4 x 256 = 1024; 4 waves/block
    cg_gemm_sig_kernel<<<256, 128, 0, stream>>>(g, w1, b1, h, C_, HID_ / 16);

    // 3) w = sigmoid(h @ w2^T + b2): tiles = 4 x 512 = 2048; 4 waves/block
    cg_gemm_sig_kernel<<<512, 128, 0, stream>>>(h, w2, b2, w_out, HID_, N2_ / 16);

    // 4) feat = bmm(w, x)/C: tiles = 64 x 13 = 832; 4 waves/block
    cg_feat_kernel<<<208, 128, 0, stream>>>(w_out, x, feat_out);
}